// GatNet_64991445123409
// MI455X (gfx1250) — compile-verified
//
#include <hip/hip_runtime.h>
#include <hip/hip_bf16.h>
#include <math.h>

typedef float v2f __attribute__((ext_vector_type(2)));
typedef float v8f __attribute__((ext_vector_type(8)));

#define HEADS 8
#define NEG_SLOPE 0.2f
#define NGRAPHS 256

// ---------------------------------------------------------------- utilities
__global__ void fill_u32(unsigned* __restrict__ p, int n, unsigned bits) {
  int i = blockIdx.x * blockDim.x + threadIdx.x;
  if (i < n) p[i] = bits;
}

__device__ __forceinline__ void atomic_max_f32(float* addr, float val) {
  // float ordering == int ordering for non-negative, reversed-unsigned for negative
  if (val >= 0.f) atomicMax((int*)addr, __float_as_int(val));
  else            atomicMin((unsigned int*)addr, __float_as_uint(val));
}

// ---------------------------------------------------------------- FP32 WMMA GEMM
// C[M,N] = A[M,K] @ B[K,N], row-major, one wave (32 lanes) per 16x16 C tile.
// V_WMMA_F32_16X16X4_F32 layout (ISA 7.12.2):
//   A 16x4:  lanes 0-15 -> rows M=0..15, K={k0,k0+1}; lanes 16-31 -> K={k0+2,k0+3}
//   B 4x16:  VGPR0: row K (lanes 0-15) / K+2 (lanes 16-31); VGPR1: K+1 / K+3
//   C/D:     VGPR i: row M=i (lanes 0-15), M=i+8 (lanes 16-31), col N = lane&15
__global__ void gemm_wmma_f32(const float* __restrict__ A, const float* __restrict__ B,
                              float* __restrict__ C, int M, int N, int K) {
  int lane = threadIdx.x & 31;
  int tm = blockIdx.x, tn = blockIdx.y;
  int l15 = lane & 15;
  int hi  = lane >> 4;          // 0 for lanes 0-15, 1 for lanes 16-31
  int mrow = tm * 16 + l15;
  int mld  = mrow < M ? mrow : (M - 1);   // clamped load row (stores are guarded)
  int ncol = tn * 16 + l15;
  v8f acc = {};
  for (int k0 = 0; k0 < K; k0 += 4) {
    int ka = k0 + (hi << 1);
    v2f a, b;
    a.x = A[(size_t)mld * K + ka];
    a.y = A[(size_t)mld * K + ka + 1];
    b.x = B[(size_t)ka * N + ncol];
    b.y = B[(size_t)(ka + 1) * N + ncol];
    acc = __builtin_amdgcn_wmma_f32_16x16x4_f32(
        /*neg_a=*/false, a, /*neg_b=*/false, b,
        /*c_mod=*/(short)0, acc, /*reuse_a=*/false, /*reuse_b=*/false);
  }
  int rbase = tm * 16 + (hi << 3);
  for (int i = 0; i < 8; ++i) {
    int r = rbase + i;
    if (r < M) C[(size_t)r * N + ncol] = acc[i];
  }
}

// ---------------------------------------------------------------- per-node alpha
__global__ void node_alpha(const float* __restrict__ h, const float* __restrict__ a_s,
                           const float* __restrict__ a_d, float* __restrict__ asrc,
                           float* __restrict__ adst, int Nn, int C) {
  int i = blockIdx.x * blockDim.x + threadIdx.x;
  if (i >= Nn * HEADS) return;
  int n = i / HEADS, hd = i % HEADS;
  const float* hp = h + (size_t)n * HEADS * C + hd * C;
  float s = 0.f, d = 0.f;
  for (int c = 0; c < C; ++c) { s += hp[c] * a_s[hd * C + c]; d += hp[c] * a_d[hd * C + c]; }
  asrc[i] = s;
  adst[i] = d;
}

// ---------------------------------------------------------------- edge passes
__device__ __forceinline__ void edge_endpoints(const int* __restrict__ ei, int NE, int e,
                                               int& s, int& d) {
  if (e < NE) { s = ei[e]; d = ei[NE + e]; }
  else        { s = e - NE; d = e - NE; }   // PyG add_self_loops
}

__global__ void edge_max(const int* __restrict__ ei, int NE, int Nn,
                         const float* __restrict__ asrc, const float* __restrict__ adst,
                         float* __restrict__ maxb) {
  int e = blockIdx.x * blockDim.x + threadIdx.x;
  int Et = NE + Nn;
  if (e >= Et) return;
  int s, d; edge_endpoints(ei, NE, e, s, d);
  for (int hd = 0; hd < HEADS; ++hd) {
    float l = asrc[s * HEADS + hd] + adst[d * HEADS + hd];
    l = l >= 0.f ? l : NEG_SLOPE * l;
    atomic_max_f32(&maxb[d * HEADS + hd], l);
  }
}

__global__ void edge_denom(const int* __restrict__ ei, int NE, int Nn,
                           const float* __restrict__ asrc, const float* __restrict__ adst,
                           const float* __restrict__ maxb, float* __restrict__ denom) {
  int e = blockIdx.x * blockDim.x + threadIdx.x;
  int Et = NE + Nn;
  if (e >= Et) return;
  int s, d; edge_endpoints(ei, NE, e, s, d);
  for (int hd = 0; hd < HEADS; ++hd) {
    float l = asrc[s * HEADS + hd] + adst[d * HEADS + hd];
    l = l >= 0.f ? l : NEG_SLOPE * l;
    float ev = __expf(l - maxb[d * HEADS + hd]);
    atomicAdd(&denom[d * HEADS + hd], ev);
  }
}

__global__ void edge_agg(const int* __restrict__ ei, int NE, int Nn,
                         const float* __restrict__ asrc, const float* __restrict__ adst,
                         const float* __restrict__ maxb, const float* __restrict__ denom,
                         const float* __restrict__ h, float* __restrict__ acc, int C) {
  int i = blockIdx.x * blockDim.x + threadIdx.x;
  int Et = NE + Nn;
  if (i >= Et * HEADS) return;
  int e = i / HEADS, hd = i % HEADS;
  int s, d; edge_endpoints(ei, NE, e, s, d);
  float l = asrc[s * HEADS + hd] + adst[d * HEADS + hd];
  l = l >= 0.f ? l : NEG_SLOPE * l;
  float ev = __expf(l - maxb[d * HEADS + hd]);
  float alpha = ev / (denom[d * HEADS + hd] + 1e-16f);
  const float* hp = h + (size_t)s * HEADS * C + hd * C;
  float* op = acc + (size_t)d * HEADS * C + hd * C;
  for (int c = 0; c < C; ++c) atomicAdd(&op[c], hp[c] * alpha);
}

// ---------------------------------------------------------------- bias + ELU
__global__ void bias_elu(float* __restrict__ acc, const float* __restrict__ bias,
                         int Nn, int F) {
  int i = blockIdx.x * blockDim.x + threadIdx.x;
  if (i >= Nn * F) return;
  float v = acc[i] + bias[i % F];
  acc[i] = v > 0.f ? v : (__expf(v) - 1.f);
}

// ---------------------------------------------------------------- pooling + head
__global__ void pool_add(const float* __restrict__ feat, const int* __restrict__ batch,
                         float* __restrict__ pooled, float* __restrict__ counts, int Nn) {
  int i = blockIdx.x * blockDim.x + threadIdx.x;
  if (i >= Nn * 128) return;
  int n = i >> 7, f = i & 127;
  int g = batch[n];
  atomicAdd(&pooled[g * 128 + f], feat[i]);
  if (f == 0) atomicAdd(&counts[g], 1.f);
}

__global__ void mlp_head(const float* __restrict__ pooled, const float* __restrict__ counts,
                         const float* __restrict__ w1, const float* __restrict__ b1,
                         const float* __restrict__ w2, const float* __restrict__ b2,
                         float* __restrict__ out, int G) {
  int g = blockIdx.x * blockDim.x + threadIdx.x;
  if (g >= G) return;
  float cnt = counts[g];
  cnt = cnt > 1.f ? cnt : 1.f;
  float inv = 1.f / cnt;
  float o = b2[0];
  for (int j = 0; j < 10; ++j) {
    float z = b1[j];
    for (int i = 0; i < 128; ++i) z += pooled[g * 128 + i] * inv * w1[i * 10 + j];
    z = z > 0.f ? z : 0.f;
    o += z * w2[j];
  }
  out[g] = o;
}

// ---------------------------------------------------------------- launcher
extern "C" void kernel_launch(void* const* d_in, const int* in_sizes, int n_in,
                              void* d_out, int out_size, void* d_ws, size_t ws_size,
                              hipStream_t stream) {
  const float* x      = (const float*)d_in[0];
  const float* W1     = (const float*)d_in[1];
  const float* a_s1   = (const float*)d_in[2];
  const float* a_d1   = (const float*)d_in[3];
  const float* b1     = (const float*)d_in[4];
  const float* W2     = (const float*)d_in[5];
  const float* a_s2   = (const float*)d_in[6];
  const float* a_d2   = (const float*)d_in[7];
  const float* b2     = (const float*)d_in[8];
  const float* W3     = (const float*)d_in[9];
  const float* a_s3   = (const float*)d_in[10];
  const float* a_d3   = (const float*)d_in[11];
  const float* b3     = (const float*)d_in[12];
  const float* fc1_w  = (const float*)d_in[13];
  const float* fc1_b  = (const float*)d_in[14];
  const float* fc2_w  = (const float*)d_in[15];
  const float* fc2_b  = (const float*)d_in[16];
  const int*   ei     = (const int*)d_in[17];
  const int*   batch  = (const int*)d_in[18];

  const int Nn = in_sizes[0] / 32;       // 50000
  const int NE = in_sizes[17] / 2;       // 800000
  const int Et = NE + Nn;

  float* ws     = (float*)d_ws;
  size_t nf     = (size_t)Nn * 128;
  float* buf0   = ws;                    // GEMM output h
  float* buf1   = buf0 + nf;             // features ping
  float* buf2   = buf1 + nf;             // features pong
  float* asrc   = buf2 + nf;             // [Nn,8]
  float* adst   = asrc + (size_t)Nn * HEADS;
  float* maxb   = adst + (size_t)Nn * HEADS;
  float* denomb = maxb + (size_t)Nn * HEADS;
  float* pooled = denomb + (size_t)Nn * HEADS;   // [256,128]
  float* counts = pooled + NGRAPHS * 128;        // [256]

  auto cdiv = [](int a, int b) { return (a + b - 1) / b; };

  auto run_layer = [&](const float* xin, int Fin, const float* W, const float* as_,
                       const float* ad_, const float* bias, int C,
                       float* hbuf, float* acc) {
    const int F = HEADS * C;
    dim3 grid(cdiv(Nn, 16), F / 16);
    gemm_wmma_f32<<<grid, 32, 0, stream>>>(xin, W, hbuf, Nn, F, Fin);
    node_alpha<<<cdiv(Nn * HEADS, 256), 256, 0, stream>>>(hbuf, as_, ad_, asrc, adst, Nn, C);
    fill_u32<<<cdiv(Nn * HEADS, 256), 256, 0, stream>>>((unsigned*)maxb, Nn * HEADS, 0xFF800000u);
    fill_u32<<<cdiv(Nn * HEADS, 256), 256, 0, stream>>>((unsigned*)denomb, Nn * HEADS, 0u);
    fill_u32<<<cdiv(Nn * F, 256), 256, 0, stream>>>((unsigned*)acc, Nn * F, 0u);
    edge_max<<<cdiv(Et, 256), 256, 0, stream>>>(ei, NE, Nn, asrc, adst, maxb);
    edge_denom<<<cdiv(Et, 256), 256, 0, stream>>>(ei, NE, Nn, asrc, adst, maxb, denomb);
    edge_agg<<<cdiv(Et * HEADS, 256), 256, 0, stream>>>(ei, NE, Nn, asrc, adst, maxb, denomb,
                                                        hbuf, acc, C);
    bias_elu<<<cdiv(Nn * F, 256), 256, 0, stream>>>(acc, bias, Nn, F);
  };

  run_layer(x,    32,  W1, a_s1, a_d1, b1, 8,  buf0, buf1);  // feat1 -> buf1 [Nn,64]
  run_layer(buf1, 64,  W2, a_s2, a_d2, b2, 16, buf0, buf2);  // feat2 -> buf2 [Nn,128]
  run_layer(buf2, 128, W3, a_s3, a_d3, b3, 16, buf0, buf1);  // feat3 -> buf1 [Nn,128]

  fill_u32<<<cdiv(NGRAPHS * 129, 256), 256, 0, stream>>>((unsigned*)pooled,
                                                         NGRAPHS * 128 + NGRAPHS, 0u);
  pool_add<<<cdiv(Nn * 128, 256), 256, 0, stream>>>(buf1, batch, pooled, counts, Nn);
  mlp_head<<<1, 256, 0, stream>>>(pooled, counts, fc1_w, fc1_b, fc2_w, fc2_b,
                                  (float*)d_out, NGRAPHS);
}